// NeighborMessageFunction_46531675685318
// MI455X (gfx1250) — compile-verified
//
#include <hip/hip_runtime.h>

typedef __attribute__((ext_vector_type(2))) float v2f;
typedef __attribute__((ext_vector_type(8))) float v8f;

#define RAW_DIM   256
#define NEIGH_DIM 128
#define MSG_DIM   128
#define N_NEIGH   20
#define M_TILE    16
#define LDA_RAW   260   // 256 + 4 pad: stride%64 == 4 -> conflict-free A reads, 16B aligned
#define LDA_128   132   // 128 + 4 pad

// Accumulate a 16x16 f32 C tile over K = 4*KTILES using V_WMMA_F32_16X16X4_F32.
// As: LDS A tile [16][LDA] (row-major, M x K). B: global weights [K][128] row-major.
// A frag layout (ISA 7.12.2): lanes 0-15 hold M=lane, {K=4kt, 4kt+1}; lanes 16-31 hold
// M=lane-16, {K=4kt+2, 4kt+3}. B frag: lanes 0-15 hold N=lane rows {K,K+1}; lanes 16-31
// hold N=lane-16 rows {K+2,K+3}.
template <int KTILES, int LDA>
__device__ __forceinline__ v8f gemm_tile(const float* __restrict__ As,
                                         const float* __restrict__ B,
                                         int ncol0, int lane) {
    v8f acc = {};
    const int mn    = lane & 15;
    const int khalf = (lane >> 4) << 1;   // 0 or 2
#pragma unroll
    for (int kt = 0; kt < KTILES; ++kt) {
        const int kb = kt * 4 + khalf;
        v2f a, b;
        a.x = As[mn * LDA + kb];
        a.y = As[mn * LDA + kb + 1];
        b.x = B[(size_t)(kb)     * MSG_DIM + ncol0 + mn];
        b.y = B[(size_t)(kb + 1) * MSG_DIM + ncol0 + mn];
        acc = __builtin_amdgcn_wmma_f32_16x16x4_f32(false, a, false, b,
                                                    (short)0, acc, false, false);
    }
    return acc;
}

__global__ __launch_bounds__(256) void neighbor_message_kernel(
    const float*     __restrict__ raw,       // [n_src, 256]
    const long long* __restrict__ neighbors, // [n_src, 20] int64
    const float*     __restrict__ memory,    // [n_nodes, 128]
    const float*     __restrict__ W1, const float* __restrict__ b1,  // [256,128],[128]
    const float*     __restrict__ W2, const float* __restrict__ b2,  // [128,128],[128]
    const float*     __restrict__ Wn, const float* __restrict__ bn,  // [128,128],[128]
    float*           __restrict__ out,       // [n_src, 128]
    int n_src)
{
    __shared__ float s_raw[M_TILE * LDA_RAW];  // 16.25 KB
    __shared__ float s_h  [M_TILE * LDA_128];  //  8.25 KB
    __shared__ float s_agg[M_TILE * LDA_128];  //  8.25 KB

    const int tid  = threadIdx.x;
    const int wave = tid >> 5;   // 0..7, owns output columns [16w, 16w+16)
    const int lane = tid & 31;
    const int m0   = blockIdx.x * M_TILE;

    // --- Stage raw_messages tile (16 x 256 floats) into LDS, 4 x B128 per thread ---
#pragma unroll
    for (int it = 0; it < 4; ++it) {
        const int j    = tid + it * 256;  // j = row*64 + col4, 1024 float4s total
        const int row  = j >> 6;
        const int col4 = j & 63;
        int grow = m0 + row; if (grow >= n_src) grow = n_src - 1;
        const float4 v = *reinterpret_cast<const float4*>(
            raw + (size_t)grow * RAW_DIM + col4 * 4);
        *reinterpret_cast<float4*>(&s_raw[row * LDA_RAW + col4 * 4]) = v;
    }

    // --- Gather + mean-pool 20 neighbor rows; wave w covers tile rows 2w, 2w+1 ---
#pragma unroll
    for (int r = 0; r < 2; ++r) {
        const int m = wave * 2 + r;
        int grow = m0 + m; if (grow >= n_src) grow = n_src - 1;
        const long long* nb = neighbors + (size_t)grow * N_NEIGH;  // wave-uniform -> s_load
        float ax = 0.f, ay = 0.f, az = 0.f, aw = 0.f;
#pragma unroll
        for (int k = 0; k < N_NEIGH; ++k) {   // 20 independent B128 gathers in flight
            const long long idx = nb[k];
            const float4 v = *reinterpret_cast<const float4*>(
                memory + (size_t)idx * NEIGH_DIM + lane * 4);
            ax += v.x; ay += v.y; az += v.z; aw += v.w;
        }
        const float inv = 1.0f / (float)N_NEIGH;
        float4 o; o.x = ax * inv; o.y = ay * inv; o.z = az * inv; o.w = aw * inv;
        *reinterpret_cast<float4*>(&s_agg[m * LDA_128 + lane * 4]) = o;
    }
    __syncthreads();

    const int ncol0 = wave * 16;
    const int n     = ncol0 + (lane & 15);     // output column of this lane
    const int moff  = (lane >> 4) * 8;         // C-tile M offset of this lane half

    // --- GEMM1: h = relu(raw @ W1 + b1), K = 256 (64 WMMAs) ---
    {
        v8f c = gemm_tile<RAW_DIM / 4, LDA_RAW>(s_raw, W1, ncol0, lane);
        const float bias = b1[n];
#pragma unroll
        for (int r8 = 0; r8 < 8; ++r8) {
            float v = c[r8] + bias;
            s_h[(moff + r8) * LDA_128 + n] = v > 0.f ? v : 0.f;
        }
    }
    __syncthreads();

    // --- GEMM2: a = relu(h @ W2 + b2); GEMM3: b = relu(agg @ Wn + bn); out = a + b ---
    v8f c2 = gemm_tile<NEIGH_DIM / 4, LDA_128>(s_h,   W2, ncol0, lane);
    v8f c3 = gemm_tile<NEIGH_DIM / 4, LDA_128>(s_agg, Wn, ncol0, lane);
    const float bias2 = b2[n];
    const float bias3 = bn[n];
#pragma unroll
    for (int r8 = 0; r8 < 8; ++r8) {
        float va = c2[r8] + bias2; va = va > 0.f ? va : 0.f;
        float vb = c3[r8] + bias3; vb = vb > 0.f ? vb : 0.f;
        const int grow = m0 + moff + r8;
        if (grow < n_src) out[(size_t)grow * MSG_DIM + n] = va + vb;
    }
}

extern "C" void kernel_launch(void* const* d_in, const int* in_sizes, int n_in,
                              void* d_out, int out_size, void* d_ws, size_t ws_size,
                              hipStream_t stream) {
    const float*     raw    = (const float*)d_in[0];
    const long long* neigh  = (const long long*)d_in[1];
    const float*     memory = (const float*)d_in[2];
    const float*     W1     = (const float*)d_in[3];
    const float*     b1     = (const float*)d_in[4];
    const float*     W2     = (const float*)d_in[5];
    const float*     b2     = (const float*)d_in[6];
    const float*     Wn     = (const float*)d_in[7];
    const float*     bn     = (const float*)d_in[8];
    float*           out    = (float*)d_out;

    const int n_src = in_sizes[0] / RAW_DIM;           // 100000
    const int grid  = (n_src + M_TILE - 1) / M_TILE;   // 6250
    neighbor_message_kernel<<<grid, 256, 0, stream>>>(
        raw, neigh, memory, W1, b1, W2, b2, Wn, bn, out, n_src);
}